// rgnn_37495064494211
// MI455X (gfx1250) — compile-verified
//
#include <hip/hip_runtime.h>
#include <math.h>

typedef float v2f __attribute__((ext_vector_type(2)));
typedef float v8f __attribute__((ext_vector_type(8)));

#define F_IN 512
#define HID 10
#define NCLS 41
#define WT_STRIDE 516  // 512 + 4 word pad: lanes map to distinct LDS banks for b64 reads

// ---------------- degree / norm ----------------
__global__ void k_init_deg(float* __restrict__ deg, int n) {
    int v = blockIdx.x * blockDim.x + threadIdx.x;
    if (v < n) deg[v] = 1.0f;  // self-loop contribution
}

__global__ void k_count_deg(const int* __restrict__ dst, float* __restrict__ deg, int E) {
    int e = blockIdx.x * blockDim.x + threadIdx.x;
    if (e < E) atomicAdd(&deg[dst[e]], 1.0f);
}

__global__ void k_dinv(const float* __restrict__ deg, float* __restrict__ dinv, int n) {
    int v = blockIdx.x * blockDim.x + threadIdx.x;
    if (v < n) dinv[v] = rsqrtf(deg[v]);  // deg >= 1 always (self-loop)
}

// ---------------- layer-1 GEMM: h = x @ W1 via V_WMMA_F32_16X16X4_F32 ----------------
// One wave (32 lanes) computes one 16-row x 16-col (cols 10..15 are zero pad) tile.
// A frag (16x4 f32): lane l: half=l>>4, m=l&15; a[j] = A[m][2*half + j]
// B frag (4x16 f32): b[j] = B[2*half + j][m]   (W1 transposed in LDS -> contiguous float2)
// D frag (16x16 f32): VGPR r, lane l -> D[r + 8*half][m]
__global__ __launch_bounds__(256) void k_gemm1_wmma(const float* __restrict__ x,
                                                    const float* __restrict__ W1,
                                                    float* __restrict__ h, int n) {
    __shared__ float wT[16 * WT_STRIDE];  // W1^T, padded to 16 cols with zeros
    for (int idx = threadIdx.x; idx < 16 * F_IN; idx += blockDim.x) {
        int nn = idx >> 9;          // col of W1 (0..15)
        int k  = idx & (F_IN - 1);  // row of W1 (0..511)
        wT[nn * WT_STRIDE + k] = (nn < HID) ? W1[k * HID + nn] : 0.0f;
    }
    __syncthreads();

    int wave = threadIdx.x >> 5;
    int lane = threadIdx.x & 31;
    int half = lane >> 4;
    int m    = lane & 15;
    int tile = blockIdx.x * 8 + wave;
    int numTiles = (n + 15) >> 4;
    if (tile >= numTiles) return;

    int row  = tile * 16 + m;
    int rowc = row < n ? row : n - 1;              // clamp loads; stores are guarded
    const float* xr = x + (size_t)rowc * F_IN;

    v8f acc = {0.f, 0.f, 0.f, 0.f, 0.f, 0.f, 0.f, 0.f};
#pragma unroll 4
    for (int k = 0; k < F_IN; k += 4) {
        v2f a = *(const v2f*)(xr + k + 2 * half);
        v2f b = *(const v2f*)(&wT[m * WT_STRIDE + k + 2 * half]);
        acc = __builtin_amdgcn_wmma_f32_16x16x4_f32(
            /*neg_a=*/false, a, /*neg_b=*/false, b,
            /*c_mod=*/(short)0, acc, /*reuse_a=*/false, /*reuse_b=*/false);
    }

    if (m < HID) {
#pragma unroll
        for (int r = 0; r < 8; r++) {
            int ro = tile * 16 + r + 8 * half;
            if (ro < n) h[(size_t)ro * HID + m] = acc[r];
        }
    }
}

// ---------------- propagation ----------------
// agg[v] = h[v] * dinv[v]^2  (self-loop term), then edges accumulate on top
__global__ void k_init_agg(const float* __restrict__ h, const float* __restrict__ dinv,
                           float* __restrict__ agg, int n) {
    int i = blockIdx.x * blockDim.x + threadIdx.x;
    if (i < n * HID) {
        int v = i / HID;
        float di = dinv[v];
        agg[i] = h[i] * di * di;
    }
}

__global__ void k_edge_prop(const int* __restrict__ src, const int* __restrict__ dst,
                            const float* __restrict__ dinv, const float* __restrict__ h,
                            float* __restrict__ agg, int E) {
    int e = blockIdx.x * blockDim.x + threadIdx.x;
    if (e >= E) return;
    int s = src[e], d = dst[e];
    float nrm = dinv[s] * dinv[d];
    const float* hs = h + (size_t)s * HID;   // 40B record: 8B aligned
    float* ad = agg + (size_t)d * HID;
#pragma unroll
    for (int j = 0; j < HID; j += 2) {
        float2 v = *(const float2*)(hs + j);
        atomicAdd(ad + j,     v.x * nrm);
        atomicAdd(ad + j + 1, v.y * nrm);
    }
}

__global__ void k_bias_relu(const float* __restrict__ agg, const float* __restrict__ b1,
                            float* __restrict__ h1, int total) {
    int i = blockIdx.x * blockDim.x + threadIdx.x;
    if (i < total) {
        float t = agg[i] + b1[i % HID];
        h1[i] = t > 0.0f ? t : 0.0f;
    }
}

// ---------------- layer-2 transform + log_softmax ----------------
// out[v] = log_softmax(agg2[v] @ W2 + b2)   (propagation commuted before W2)
__global__ void k_final(const float* __restrict__ agg, const float* __restrict__ W2,
                        const float* __restrict__ b2, float* __restrict__ out, int n) {
    int v = blockIdx.x * blockDim.x + threadIdx.x;
    if (v >= n) return;
    float g[HID];
#pragma unroll
    for (int k = 0; k < HID; k++) g[k] = agg[(size_t)v * HID + k];
    float z[NCLS];
    float mx = -INFINITY;
#pragma unroll
    for (int c = 0; c < NCLS; c++) {
        float a = b2[c];
#pragma unroll
        for (int k = 0; k < HID; k++) a += g[k] * W2[k * NCLS + c];
        z[c] = a;
        mx = fmaxf(mx, a);
    }
    float sum = 0.0f;
#pragma unroll
    for (int c = 0; c < NCLS; c++) sum += expf(z[c] - mx);
    float lse = logf(sum) + mx;
    float* o = out + (size_t)v * NCLS;
#pragma unroll
    for (int c = 0; c < NCLS; c++) o[c] = z[c] - lse;
}

extern "C" void kernel_launch(void* const* d_in, const int* in_sizes, int n_in,
                              void* d_out, int out_size, void* d_ws, size_t ws_size,
                              hipStream_t stream) {
    const float* x    = (const float*)d_in[0];
    const float* W1   = (const float*)d_in[1];
    const float* b1   = (const float*)d_in[2];
    const float* W2   = (const float*)d_in[3];
    const float* b2   = (const float*)d_in[4];
    const int*   esrc = (const int*)d_in[5];
    const int*   edst = (const int*)d_in[6];
    float* out = (float*)d_out;

    int n = in_sizes[0] / F_IN;   // 100000
    int E = in_sizes[5];          // 3200000

    float* deg  = (float*)d_ws;               // [n]
    float* dinv = deg + n;                    // [n]
    float* bufA = dinv + n;                   // [n*HID]  h / h1
    float* bufB = bufA + (size_t)n * HID;     // [n*HID]  agg

    dim3 blk(256);
    dim3 gN((n + 255) / 256);
    dim3 gE((E + 255) / 256);
    dim3 gNH((n * HID + 255) / 256);
    int numTiles = (n + 15) / 16;
    dim3 gG((numTiles + 7) / 8);  // 8 waves / 256-thread block, one 16-row tile per wave

    // degree + symmetric norm
    k_init_deg<<<gN, blk, 0, stream>>>(deg, n);
    k_count_deg<<<gE, blk, 0, stream>>>(edst, deg, E);
    k_dinv<<<gN, blk, 0, stream>>>(deg, dinv, n);

    // layer 1: dense transform (WMMA f32), then normalized propagation
    k_gemm1_wmma<<<gG, blk, 0, stream>>>(x, W1, bufA, n);
    k_init_agg<<<gNH, blk, 0, stream>>>(bufA, dinv, bufB, n);
    k_edge_prop<<<gE, blk, 0, stream>>>(esrc, edst, dinv, bufA, bufB, E);
    k_bias_relu<<<gNH, blk, 0, stream>>>(bufB, b1, bufA, n * HID);

    // layer 2: propagate 10-dim features first (P and W2 commute), then W2+b2+log_softmax
    k_init_agg<<<gNH, blk, 0, stream>>>(bufA, dinv, bufB, n);
    k_edge_prop<<<gE, blk, 0, stream>>>(esrc, edst, dinv, bufA, bufB, E);
    k_final<<<gN, blk, 0, stream>>>(bufB, W2, b2, out, n);
}